// Sampler_84335977825021
// MI455X (gfx1250) — compile-verified
//
#include <hip/hip_runtime.h>
#include <math.h>

// ---------------------------------------------------------------------------
// Per-row sampler: temperature softmax + top-k/top-p/min-p + multinomial draw.
// One workgroup (1024 threads = 32 wave32) per batch row. Two streaming passes:
//   pass 1 (HBM): online-softmax max+sum, 2048-bucket histogram of logit
//                 order-keys (monotone with prob -> thresholds need no exp)
//   pass 2 (L2):  async-staged (global_load_async_to_lds_b64) compaction of
//                 the >=2048 top candidates; exp only for survivors.
// Candidates packed as u64 keys (e_bits<<32 | ~idx): one-compare bitonic sort
// with register-resident j<=2 stages, prefix sums written into the key high
// word in place, then prefix-length selection n = min(top_k, n_p, n_m) and
// inverse-CDF sampling.
// ---------------------------------------------------------------------------

#define BLOCK 1024
#define CAP   4096           // candidate capacity (power of two for bitonic)
#define NBKT  2048           // key>>21 covers the full 32-bit key space
#define CHUNK 2048           // floats per async-staged chunk (8KB buffer)
#define PT    (CAP / BLOCK)  // 4 owned elements per thread

typedef unsigned long long u64;

__device__ __forceinline__ int imin(int a, int b) { return a < b ? a : b; }

// Order-preserving uint key for float (total order, handles negatives).
__device__ __forceinline__ unsigned fkey(float x) {
    unsigned u = __float_as_uint(x);
    unsigned mask = ((unsigned)((int)u >> 31)) | 0x80000000u;
    return u ^ mask;
}

__device__ __forceinline__ void cswap(u64& a, u64& b, bool desc) {
    bool sw = desc ? (a < b) : (a > b);
    if (sw) { u64 t = a; a = b; b = t; }
}

__global__ __launch_bounds__(BLOCK)
void sampler_topk_topp_minp_kernel(const float* __restrict__ logits,
                                   const float* __restrict__ temps,
                                   const int*   __restrict__ topks,
                                   const float* __restrict__ topps,
                                   const float* __restrict__ minps,
                                   const float* __restrict__ us,
                                   float* __restrict__ out,
                                   int V)
{
    const int b   = blockIdx.x;
    const int tid = threadIdx.x;
    const float* __restrict__ row = logits + (size_t)b * (size_t)V;

    __shared__ __align__(16) u64      s_kv[CAP];        // 32KB: keys -> F in hi word
    __shared__ float                  s_part[BLOCK];    // 4KB reduction/scan partials
    __shared__ __align__(16) unsigned s_scr[2 * CHUNK]; // 16KB: hist | async stage bufs
    __shared__ float    s_m, s_S, s_r;
    __shared__ unsigned s_tbits, s_cnt, s_np, s_samp, s_nmc;
    __shared__ int      s_n;

    unsigned* kv32  = (unsigned*)s_kv;    // [2i]=~idx (lo), [2i+1]=val/F bits (hi)
    float*    s_fvl = (float*)s_kv;       // pass-1 scratch for running maxes

    const float T    = temps[b];
    const float invT = 1.0f / T;

    for (int i = tid; i < NBKT; i += BLOCK) s_scr[i] = 0u;
    if (tid == 0) { s_tbits = 0u; s_cnt = 0u; s_np = 0u; s_samp = 0u; s_nmc = 0u; }
    __syncthreads();

    // ---------------- Pass 1 (HBM): online max+sum, key histogram ----------
    float lm = -INFINITY, ls = 0.0f;
    const int nvec = ((V & 3) == 0) ? (V >> 2) : 0;
    const float4* __restrict__ rowv = reinterpret_cast<const float4*>(row);
    for (int j = tid; j < nvec; j += BLOCK) {
        __builtin_prefetch((const void*)(rowv + j + 8 * BLOCK), 0, 3); // global_prefetch_b8
        float4 x = rowv[j];
        float xs[4] = { x.x, x.y, x.z, x.w };
        #pragma unroll
        for (int q = 0; q < 4; ++q) {
            float l = xs[q];
            if (l > lm) { ls *= __expf((lm - l) * invT); lm = l; }  // first iter: 0*0=0
            ls += __expf((l - lm) * invT);
            atomicAdd(&s_scr[fkey(l) >> 21], 1u);
        }
    }
    for (int i = (nvec << 2) + tid; i < V; i += BLOCK) {
        float l = row[i];
        if (l > lm) { ls *= __expf((lm - l) * invT); lm = l; }
        ls += __expf((l - lm) * invT);
        atomicAdd(&s_scr[fkey(l) >> 21], 1u);
    }
    // deterministic pairwise combine of (max, rescaled sum)
    s_fvl[tid] = lm; s_part[tid] = ls;
    __syncthreads();
    for (int off = BLOCK >> 1; off > 0; off >>= 1) {
        if (tid < off) {
            float m1 = s_fvl[tid],        s1 = s_part[tid];
            float m2 = s_fvl[tid + off],  s2 = s_part[tid + off];
            float M  = fmaxf(m1, m2);
            float c1 = (m1 > -INFINITY) ? __expf((m1 - M) * invT) : 0.0f;
            float c2 = (m2 > -INFINITY) ? __expf((m2 - M) * invT) : 0.0f;
            s_fvl[tid]  = M;
            s_part[tid] = s1 * c1 + s2 * c2;
        }
        __syncthreads();
    }
    if (tid == 0) { s_m = s_fvl[0]; s_S = s_part[0]; }
    __syncthreads();
    const float m = s_m;
    const float S = s_S;

    // ------------- Parallel suffix-scan of histogram -> key threshold ------
    {
        const unsigned NEED = (unsigned)imin(V, 2048);
        const int r0 = 2 * tid, r1 = 2 * tid + 1;          // reversed positions
        const int b0 = NBKT - 1 - r0, b1 = NBKT - 1 - r1;  // bucket ids
        unsigned a0 = s_scr[b0], a1 = s_scr[b1];
        s_part[tid] = (float)(a0 + a1);                    // counts exact in f32
        __syncthreads();
        for (int off = 1; off < BLOCK; off <<= 1) {
            float v   = s_part[tid];
            float add = (tid >= off) ? s_part[tid - off] : 0.0f;
            __syncthreads();
            s_part[tid] = v + add;
            __syncthreads();
        }
        float exclf = (tid > 0) ? s_part[tid - 1] : 0.0f;
        unsigned c0 = (unsigned)exclf + a0;                // topcount(b0)
        unsigned c1 = c0 + a1;                             // topcount(b1)
        s_scr[b0] = c0; s_scr[b1] = c1;
        __syncthreads();
        #pragma unroll
        for (int q = 0; q < 2; ++q) {
            int bb = q ? b1 : b0;
            unsigned tc  = s_scr[bb];
            unsigned tcn = (bb == NBKT - 1) ? 0u : s_scr[bb + 1];
            if (tc >= NEED && tcn < NEED) {                // unique crossing thread
                int tb = (tc <= CAP || tcn == 0u) ? bb : bb + 1;
                s_tbits = ((unsigned)tb) << 21;
            }
        }
        __syncthreads();
    }
    const unsigned tbits = s_tbits;

    // -------- Pass 2 (L2): async-staged compaction of candidates -----------
    // Each lane stages and consumes only its own 8 bytes per chunk, so only
    // per-wave ASYNCcnt ordering is needed (no workgroup barriers).
    const int nfull = ((V & 3) == 0) ? (V / CHUNK) : 0;    // b64 alignment safe
    const unsigned long long gbase = (unsigned long long)(size_t)row;
    auto issue = [&](int buf, int c) {
        unsigned lds_addr = (unsigned)(size_t)((float*)s_scr + buf * CHUNK) + 8u * (unsigned)tid;
        unsigned goff     = (unsigned)(c * CHUNK) * 4u + 8u * (unsigned)tid;
        asm volatile("global_load_async_to_lds_b64 %0, %1, %2 offset:0"
                     :: "v"(lds_addr), "v"(goff), "s"(gbase) : "memory");
    };
    if (nfull > 0) issue(0, 0);
    for (int c = 0; c < nfull; ++c) {
        const int buf = c & 1;
        if (c + 1 < nfull) {
            issue(buf ^ 1, c + 1);
            asm volatile("s_wait_asynccnt 0x1" ::: "memory"); // oldest chunk landed
        } else {
            asm volatile("s_wait_asynccnt 0x0" ::: "memory");
        }
        const float2 lv = *reinterpret_cast<const float2*>(
            (const char*)((const float*)s_scr + buf * CHUNK) + 8u * (unsigned)tid);
        const int gi = c * CHUNK + 2 * tid;
        #pragma unroll
        for (int q = 0; q < 2; ++q) {
            float l = q ? lv.y : lv.x;
            if (fkey(l) >= tbits) {
                float e = __expf((l - m) * invT);          // exp only for survivors
                unsigned p = atomicAdd(&s_cnt, 1u);
                if (p < CAP)
                    s_kv[p] = ((u64)__float_as_uint(e) << 32)
                            | (u64)(unsigned)(~(unsigned)(gi + q));
            }
        }
    }
    for (int i = nfull * CHUNK + tid; i < V; i += BLOCK) { // tail, plain loads
        float l = row[i];
        if (fkey(l) >= tbits) {
            float e = __expf((l - m) * invT);
            unsigned p = atomicAdd(&s_cnt, 1u);
            if (p < CAP)
                s_kv[p] = ((u64)__float_as_uint(e) << 32)
                        | (u64)(unsigned)(~(unsigned)i);
        }
    }
    __syncthreads();
    const int cnt_eff = imin((int)s_cnt, CAP);

    // min-p count over candidates (exact: binds only when cutoff > threshold)
    {
        const float minp = minps[b];
        unsigned lc = 0u;
        for (int i = tid; i < cnt_eff; i += BLOCK)
            if (__uint_as_float(kv32[2 * i + 1]) >= minp) ++lc;
        if (lc) atomicAdd(&s_nmc, lc);
    }
    for (int i = tid; i < CAP; i += BLOCK)                 // pad key 0 sorts last
        if (i >= cnt_eff) s_kv[i] = 0ull;
    __syncthreads();

    // -------- Bitonic sort on packed u64 keys (desc = val desc, idx asc) ---
    const int e0 = tid * PT;                               // owned block [e0, e0+4)
    {   // k=2 and k=4 stages entirely in registers
        u64 k0 = s_kv[e0 + 0], k1 = s_kv[e0 + 1], k2 = s_kv[e0 + 2], k3 = s_kv[e0 + 3];
        cswap(k0, k1, true);            // k=2: (e0&2)==0 always
        cswap(k2, k3, false);           //      (e0+2)&2 != 0
        bool d4 = ((e0 & 4) == 0);      // k=4
        cswap(k0, k2, d4); cswap(k1, k3, d4);
        cswap(k0, k1, d4); cswap(k2, k3, d4);
        s_kv[e0 + 0] = k0; s_kv[e0 + 1] = k1; s_kv[e0 + 2] = k2; s_kv[e0 + 3] = k3;
    }
    __syncthreads();
    for (int k = 8; k <= CAP; k <<= 1) {
        for (int j = k >> 1; j >= 4; j >>= 1) {            // cross-thread stages
            for (int i = tid; i < CAP; i += BLOCK) {
                int ix = i ^ j;
                if (ix > i) {
                    bool desc = ((i & k) == 0);
                    u64 a = s_kv[i], c = s_kv[ix];
                    bool sw = desc ? (a < c) : (a > c);
                    if (sw) { s_kv[i] = c; s_kv[ix] = a; }
                }
            }
            __syncthreads();
        }
        {   // j=2, j=1 stages in registers (direction uniform over owned block)
            u64 k0 = s_kv[e0 + 0], k1 = s_kv[e0 + 1], k2 = s_kv[e0 + 2], k3 = s_kv[e0 + 3];
            bool d = ((e0 & k) == 0);
            cswap(k0, k2, d); cswap(k1, k3, d);
            cswap(k0, k1, d); cswap(k2, k3, d);
            s_kv[e0 + 0] = k0; s_kv[e0 + 1] = k1; s_kv[e0 + 2] = k2; s_kv[e0 + 3] = k3;
        }
        __syncthreads();
    }

    // -------- Inclusive prefix sum F written into key high words -----------
    {
        float v0 = __uint_as_float(kv32[2 * (e0 + 0) + 1]);
        float v1 = __uint_as_float(kv32[2 * (e0 + 1) + 1]);
        float v2 = __uint_as_float(kv32[2 * (e0 + 2) + 1]);
        float v3 = __uint_as_float(kv32[2 * (e0 + 3) + 1]);
        float f0 = v0, f1 = f0 + v1, f2 = f1 + v2, f3 = f2 + v3;
        s_part[tid] = f3;
        __syncthreads();
        for (int off = 1; off < BLOCK; off <<= 1) {
            float v   = s_part[tid];
            float add = (tid >= off) ? s_part[tid - off] : 0.0f;
            __syncthreads();
            s_part[tid] = v + add;
            __syncthreads();
        }
        float prev = (tid > 0) ? s_part[tid - 1] : 0.0f;
        kv32[2 * (e0 + 0) + 1] = __float_as_uint(f0 + prev);
        kv32[2 * (e0 + 1) + 1] = __float_as_uint(f1 + prev);
        kv32[2 * (e0 + 2) + 1] = __float_as_uint(f2 + prev);
        kv32[2 * (e0 + 3) + 1] = __float_as_uint(f3 + prev);
    }
    __syncthreads();
    // hi(kv[j]) = F(j+1): cumulative mass of top (j+1); lo(kv[j]) = ~token_id

    // ---------------- Select prefix length n and sample --------------------
    const float P = topps[b] * S;          // top-p bound on unnormalized cumsum
    {
        int lim = imin(V, 2048);
        unsigned lc = 0u;
        for (int jd = tid; jd < lim; jd += BLOCK) {
            float F = (jd == 0) ? 0.0f : __uint_as_float(kv32[2 * (jd - 1) + 1]);
            if (F <= P) ++lc;
        }
        atomicAdd(&s_np, lc);
    }
    __syncthreads();
    if (tid == 0) {
        int n = topks[b];
        n = imin(n, (int)s_np);
        n = imin(n, (int)s_nmc);
        n = imin(n, cnt_eff);
        if (n < 1) n = 1;
        s_n = n;
        float K = __uint_as_float(kv32[2 * (n - 1) + 1]);  // total kept mass
        s_r = us[b] * K;
    }
    __syncthreads();
    {
        const int   n = s_n;
        const float r = s_r;
        unsigned lc = 0u;
        for (int jd = tid; jd < n; jd += BLOCK)
            if (__uint_as_float(kv32[2 * jd + 1]) < r) ++lc;   // #{F(j) < r}
        atomicAdd(&s_samp, lc);
    }
    __syncthreads();
    if (tid == 0) {
        int s = imin((int)s_samp, s_n - 1);
        if (s < 0) s = 0;
        unsigned token = ~kv32[2 * s];
        out[b] = (float)token;
    }
}

extern "C" void kernel_launch(void* const* d_in, const int* in_sizes, int n_in,
                              void* d_out, int out_size, void* d_ws, size_t ws_size,
                              hipStream_t stream) {
    const float* logits = (const float*)d_in[0];
    const float* temps  = (const float*)d_in[1];
    const int*   topks  = (const int*)  d_in[2];
    const float* topps  = (const float*)d_in[3];
    const float* minps  = (const float*)d_in[4];
    const float* us     = (const float*)d_in[5];
    float*       out    = (float*)d_out;

    const int B = in_sizes[1];
    const int V = in_sizes[0] / B;

    hipLaunchKernelGGL(sampler_topk_topp_minp_kernel,
                       dim3(B), dim3(BLOCK), 0, stream,
                       logits, temps, topks, topps, minps, us, out, V);
    (void)d_ws; (void)ws_size; (void)n_in; (void)out_size;
}